// GNNmodel_15564961481429
// MI455X (gfx1250) — compile-verified
//
#include <hip/hip_runtime.h>
#include <hip/hip_bf16.h>

#define N_NODES  100000
#define N_EDGES  3200000
#define N_GRAPHS 64

typedef __attribute__((ext_vector_type(2))) float v2f;
typedef __attribute__((ext_vector_type(8))) float v8f;

__device__ __forceinline__ float silu_f(float x) {
    return x / (1.0f + __expf(-x));
}

// ---------------- utility ----------------
__global__ void k_zero(float* __restrict__ p, int n) {
    int i = blockIdx.x * blockDim.x + threadIdx.x;
    if (i < n) p[i] = 0.0f;
}

// deg[dst] += ew  (deg buffer pre-zeroed)
__global__ void k_deg_accum(const int* __restrict__ dst, const float* __restrict__ ew,
                            float* __restrict__ deg) {
    int e = blockIdx.x * blockDim.x + threadIdx.x;
    if (e < N_EDGES) atomicAdd(&deg[dst[e]], ew[e]);
}

// in-place: dinv[i] = rsqrt(deg[i] + 1)
__global__ void k_deg_finish(float* __restrict__ deg_dinv) {
    int i = blockIdx.x * blockDim.x + threadIdx.x;
    if (i < N_NODES) deg_dinv[i] = rsqrtf(deg_dinv[i] + 1.0f);
}

// norm[e] = dinv[src]*ew*dinv[dst]
__global__ void k_edge_norm(const int* __restrict__ src, const int* __restrict__ dst,
                            const float* __restrict__ ew, const float* __restrict__ dinv,
                            float* __restrict__ norm) {
    int e = blockIdx.x * blockDim.x + threadIdx.x;
    if (e < N_EDGES) norm[e] = dinv[src[e]] * ew[e] * dinv[dst[e]];
}

// ---------------- WMMA f32 GEMM: xw = h @ W ; agg = xw*dinv^2 + bias ----------------
// One wave computes a 16x16 output tile, K-loop in steps of 4 via v_wmma_f32_16x16x4_f32.
// N_NODES % 16 == 0, K % 4 == 0, M % 16 == 0 (holds for all three layers).
__global__ void k_gemm_wmma(const float* __restrict__ h, const float* __restrict__ W,
                            const float* __restrict__ bias, const float* __restrict__ dinv,
                            float* __restrict__ xw, float* __restrict__ agg,
                            int K, int M) {
    const int nColTiles = M >> 4;
    const int nTiles    = (N_NODES >> 4) * nColTiles;
    const int wave      = (blockIdx.x * blockDim.x + threadIdx.x) >> 5;
    if (wave >= nTiles) return;              // uniform per wave -> EXEC stays all-1s

    const int colTile = wave % nColTiles;
    const int rowTile = wave / nColTiles;
    const int lane    = threadIdx.x & 31;
    const int half    = lane >> 4;           // which 16-lane half
    const int l16     = lane & 15;
    const int rowBase = rowTile << 4;
    const int colBase = colTile << 4;

    // A fragment rows: lane l16 owns matrix row (rowBase+l16); half selects K offset {0,1} vs {2,3}
    const float* hrow = h + (size_t)(rowBase + l16) * K;

    v8f c = {};
    for (int k0 = 0; k0 < K; k0 += 4) {
        const int ka = k0 + (half << 1);
        v2f a, b;
        a.x = hrow[ka];
        a.y = hrow[ka + 1];
        b.x = W[(size_t)ka * M + colBase + l16];
        b.y = W[(size_t)(ka + 1) * M + colBase + l16];
        c = __builtin_amdgcn_wmma_f32_16x16x4_f32(
                /*neg_a=*/false, a, /*neg_b=*/false, b,
                /*c_mod=*/(short)0, c, /*reuse_a=*/false, /*reuse_b=*/false);
    }

    // D layout: VGPR v -> row = rowBase + v + 8*half, col = colBase + l16
    const int col = colBase + l16;
    const float bv = bias[col];
#pragma unroll
    for (int v = 0; v < 8; ++v) {
        const int row  = rowBase + v + (half << 3);
        const float d  = c[v];
        const size_t i = (size_t)row * M + col;
        const float di = dinv[row];
        xw[i]  = d;
        agg[i] = d * (di * di) + bv;         // self-loop (1/deg) contribution + bias
    }
}

// ---------------- edge scatter: agg[dst] += xw[src] * norm, float4-vectorized ----------------
__global__ void k_scatter(const float* __restrict__ xw, const float* __restrict__ norm,
                          const int* __restrict__ src, const int* __restrict__ dst,
                          float* __restrict__ agg, int M) {
    const int chunks = M >> 2;
    const long long tid   = (long long)blockIdx.x * blockDim.x + threadIdx.x;
    const long long total = (long long)N_EDGES * chunks;
    if (tid >= total) return;
    const int e = (int)(tid / chunks);
    const int c = ((int)(tid % chunks)) << 2;
    const int s = src[e];
    const int d = dst[e];
    const float w = norm[e];
    const float4 v = *(const float4*)(xw + (size_t)s * M + c);
    float* ap = agg + (size_t)d * M + c;
    atomicAdd(ap + 0, v.x * w);
    atomicAdd(ap + 1, v.y * w);
    atomicAdd(ap + 2, v.z * w);
    atomicAdd(ap + 3, v.w * w);
}

// ---------------- elementwise SiLU ----------------
__global__ void k_silu(const float* __restrict__ in, float* __restrict__ out, int n) {
    int i = blockIdx.x * blockDim.x + threadIdx.x;
    if (i < n) out[i] = silu_f(in[i]);
}

// ---------------- fused SiLU + global mean-pool accumulate (layer 3, M=256) ----------------
__global__ void k_pool(const float* __restrict__ agg, const int* __restrict__ batch,
                       float* __restrict__ sums, float* __restrict__ cnt) {
    const long long tid   = (long long)blockIdx.x * blockDim.x + threadIdx.x;
    const long long total = (long long)N_NODES * 64;   // 64 float4 chunks per node
    if (tid >= total) return;
    const int i = (int)(tid >> 6);
    const int c = ((int)(tid & 63)) << 2;
    const int g = batch[i];
    const float* ip = agg + (size_t)i * 256 + c;
    float* sp = sums + (size_t)g * 256 + c;
    atomicAdd(sp + 0, silu_f(ip[0]));
    atomicAdd(sp + 1, silu_f(ip[1]));
    atomicAdd(sp + 2, silu_f(ip[2]));
    atomicAdd(sp + 3, silu_f(ip[3]));
    if (c == 0) atomicAdd(&cnt[g], 1.0f);
}

__global__ void k_pooldiv(float* __restrict__ sums, const float* __restrict__ cnt) {
    int i = blockIdx.x * blockDim.x + threadIdx.x;
    if (i < N_GRAPHS * 256) {
        const float cn = cnt[i >> 8];
        sums[i] = sums[i] / fmaxf(cn, 1.0f);
    }
}

// ---------------- tiny MLP head layer: out[g,m] = act(in[g,:]@W[:,m] + b[m]) ----------------
__global__ void k_mlp(const float* __restrict__ in, const float* __restrict__ W,
                      const float* __restrict__ b, float* __restrict__ out,
                      int K, int M, int act) {
    const int g = blockIdx.x;
    const float* ip = in + (size_t)g * K;
    for (int m = threadIdx.x; m < M; m += blockDim.x) {
        float acc = b[m];
        for (int k = 0; k < K; ++k) acc += ip[k] * W[(size_t)k * M + m];
        out[(size_t)g * M + m] = act ? silu_f(acc) : acc;
    }
}

// ---------------- host side ----------------
static inline int cdiv(long long a, int b) { return (int)((a + b - 1) / b); }

extern "C" void kernel_launch(void* const* d_in, const int* in_sizes, int n_in,
                              void* d_out, int out_size, void* d_ws, size_t ws_size,
                              hipStream_t stream) {
    const float* x    = (const float*)d_in[0];
    const int*   eidx = (const int*)d_in[1];
    const int*   src  = eidx;
    const int*   dst  = eidx + N_EDGES;
    const float* ew   = (const float*)d_in[2];
    const int*   batch= (const int*)d_in[3];
    const float* W1 = (const float*)d_in[4];  const float* b1 = (const float*)d_in[5];
    const float* W2 = (const float*)d_in[6];  const float* b2 = (const float*)d_in[7];
    const float* W3 = (const float*)d_in[8];  const float* b3 = (const float*)d_in[9];
    const float* L1 = (const float*)d_in[10]; const float* c1 = (const float*)d_in[11];
    const float* L2 = (const float*)d_in[12]; const float* c2 = (const float*)d_in[13];
    const float* L3 = (const float*)d_in[14]; const float* c3 = (const float*)d_in[15];
    float* out = (float*)d_out;

    // workspace layout (floats); all region sizes are multiples of 16B
    float* ws = (float*)d_ws;
    size_t off = 0;
    float* norm = ws + off; off += (size_t)N_EDGES;          // 3.2M
    float* dinv = ws + off; off += (size_t)N_NODES;          // 100k (deg, then rsqrt(deg+1))
    float* bufXW  = ws + off; off += (size_t)N_NODES * 256;  // 25.6M
    float* bufAGG = ws + off; off += (size_t)N_NODES * 256;  // 25.6M
    float* bufH   = ws + off; off += (size_t)N_NODES * 64;   // 6.4M
    float* sums = ws + off; off += (size_t)N_GRAPHS * 256;
    float* cnt  = ws + off; off += (size_t)N_GRAPHS;
    float* z1   = ws + off; off += (size_t)N_GRAPHS * 128;
    float* z2   = ws + off; off += (size_t)N_GRAPHS * 64;

    const int T = 256;

    // normalization precompute (shared by all 3 layers)
    k_zero<<<cdiv(N_NODES, T), T, 0, stream>>>(dinv, N_NODES);
    k_zero<<<cdiv(N_GRAPHS * 256 + N_GRAPHS, T), T, 0, stream>>>(sums, N_GRAPHS * 256 + N_GRAPHS);
    k_deg_accum<<<cdiv(N_EDGES, T), T, 0, stream>>>(dst, ew, dinv);
    k_deg_finish<<<cdiv(N_NODES, T), T, 0, stream>>>(dinv);
    k_edge_norm<<<cdiv(N_EDGES, T), T, 0, stream>>>(src, dst, ew, dinv, norm);

    // ---- layer 1: 16 -> 16 ----
    {
        const int K = 16, M = 16;
        const long long waves = (long long)(N_NODES / 16) * (M / 16);
        k_gemm_wmma<<<cdiv(waves * 32, T), T, 0, stream>>>(x, W1, b1, dinv, bufXW, bufAGG, K, M);
        k_scatter<<<cdiv((long long)N_EDGES * (M / 4), T), T, 0, stream>>>(bufXW, norm, src, dst, bufAGG, M);
        k_silu<<<cdiv((long long)N_NODES * M, T), T, 0, stream>>>(bufAGG, bufH, N_NODES * M);
    }
    // ---- layer 2: 16 -> 64 ----
    {
        const int K = 16, M = 64;
        const long long waves = (long long)(N_NODES / 16) * (M / 16);
        k_gemm_wmma<<<cdiv(waves * 32, T), T, 0, stream>>>(bufH, W2, b2, dinv, bufXW, bufAGG, K, M);
        k_scatter<<<cdiv((long long)N_EDGES * (M / 4), T), T, 0, stream>>>(bufXW, norm, src, dst, bufAGG, M);
        k_silu<<<cdiv((long long)N_NODES * M, T), T, 0, stream>>>(bufAGG, bufH, N_NODES * M);
    }
    // ---- layer 3: 64 -> 256 (SiLU fused into pooling) ----
    {
        const int K = 64, M = 256;
        const long long waves = (long long)(N_NODES / 16) * (M / 16);
        k_gemm_wmma<<<cdiv(waves * 32, T), T, 0, stream>>>(bufH, W3, b3, dinv, bufXW, bufAGG, K, M);
        k_scatter<<<cdiv((long long)N_EDGES * (M / 4), T), T, 0, stream>>>(bufXW, norm, src, dst, bufAGG, M);
        k_pool<<<cdiv((long long)N_NODES * 64, T), T, 0, stream>>>(bufAGG, batch, sums, cnt);
        k_pooldiv<<<cdiv(N_GRAPHS * 256, T), T, 0, stream>>>(sums, cnt);
    }

    // ---- MLP head ----
    k_mlp<<<N_GRAPHS, 128, 0, stream>>>(sums, L1, c1, z1, 256, 128, 1);
    k_mlp<<<N_GRAPHS, 64, 0, stream>>>(z1, L2, c2, z2, 128, 64, 1);
    k_mlp<<<N_GRAPHS, 32, 0, stream>>>(z2, L3, c3, out, 64, 3, 0);
}